// ParallelExperts_23708219474100
// MI455X (gfx1250) — compile-verified
//
#include <hip/hip_runtime.h>
#include <stdint.h>

// Grouped GEMM (MoE experts) in exact fp32 using CDNA5 V_WMMA_F32_16X16X4_F32,
// with all global->LDS staging on the gfx1250 async engine
// (GLOBAL_LOAD_ASYNC_TO_LDS_B128/B32, fenced by s_wait_asynccnt).

typedef float v2f __attribute__((ext_vector_type(2)));
typedef float v8f __attribute__((ext_vector_type(8)));

#define K_IN   1024
#define K_OUT  4096
#define BM     128
#define BN     128
#define BK     16
#define STR_A  20    // BK+4: b128-aligned rows, conflict-free b64 frag loads
#define STR_B  18    // BK+2: b64-aligned rows, conflict-free b64 frag loads

__global__ __launch_bounds__(256)
void moe_grouped_gemm_f32_wmma(const float* __restrict__ x,
                               const int*   __restrict__ expert_size,
                               const float* __restrict__ w,
                               float*       __restrict__ out,
                               int E)
{
    // A stored M-major [m][k], B stored N-major [n][k] (transposed by the
    // async scatter) so every WMMA fragment is one aligned ds_load_b64.
    __shared__ float ldsA[2][BM][STR_A];
    __shared__ float ldsB[2][BN][STR_B];

    const int tid   = threadIdx.x;
    const int lane  = tid & 31;        // wave32
    const int wid   = tid >> 5;        // 8 waves per block
    const int ml    = lane & 15;       // M/N index within 16
    const int khalf = lane >> 4;       // 0: K pair {0,1}, 1: K pair {2,3}
    const int koff  = khalf * 2;

    const int row0 = blockIdx.x * BM;
    const int col0 = blockIdx.y * BN;

    // Find this row-block's expert via prefix scan of expert_size.
    // (Tiles assumed not to straddle expert boundaries: sizes are multiples of BM.)
    int e = 0;
    {
        int acc = 0;
        for (int i = 0; i < E; ++i) {
            int s = expert_size[i];
            if (row0 < acc + s) { e = i; break; }
            acc += s;
        }
    }
    const float* __restrict__ we = w + (size_t)e * K_IN * K_OUT;

    // Wave sub-tile: 64(M) x 32(N) -> 4x2 grid of 16x16 WMMA tiles.
    const int m0 = (wid & 1) * 64;
    const int n0 = (wid >> 1) * 32;

    v8f acc[4][2] = {};

    // LDS byte offsets for the async destination VGPRs (workgroup-relative).
    const unsigned ldsA_base = (unsigned)(uintptr_t)&ldsA[0][0][0];
    const unsigned ldsB_base = (unsigned)(uintptr_t)&ldsB[0][0][0];

    // Async staging indices.
    const int a_r = tid >> 2;          // 0..63 (A row; +64 on 2nd pass)
    const int a_c = (tid & 3) * 4;     // 0,4,8,12 (A k column, 16B chunk)

    auto asyncLoadTile = [&](int kt, int buf) {
        // A tile: 128x16, two b128 async copies per thread (direct copy).
        const float* ap = x + (size_t)row0 * K_IN + kt * BK;
#pragma unroll
        for (int i = 0; i < 2; ++i) {
            const int m = a_r + i * 64;
            unsigned dst = ldsA_base + (unsigned)(((buf * BM + m) * STR_A + a_c) * 4);
            unsigned long long src =
                (unsigned long long)(uintptr_t)(ap + (size_t)m * K_IN + a_c);
            asm volatile("global_load_async_to_lds_b128 %0, %1, off"
                         :: "v"(dst), "v"(src) : "memory");
        }
        // B tile: 16x128, eight b32 async copies per thread; coalesced global
        // reads, per-lane LDS scatter performs the [k][n] -> [n][k] transpose.
        const float* bp = we + (size_t)(kt * BK) * K_OUT + col0;
#pragma unroll
        for (int i = 0; i < 8; ++i) {
            const int g = tid + i * 256;   // 0..2047
            const int k = g >> 7;          // 0..15
            const int n = g & 127;         // 0..127
            unsigned dst = ldsB_base + (unsigned)(((buf * BN + n) * STR_B + k) * 4);
            unsigned long long src =
                (unsigned long long)(uintptr_t)(bp + (size_t)k * K_OUT + n);
            asm volatile("global_load_async_to_lds_b32 %0, %1, off"
                         :: "v"(dst), "v"(src) : "memory");
        }
    };

    const int NKT = K_IN / BK;  // 64 K-tiles

    asyncLoadTile(0, 0);
    asm volatile("s_wait_asynccnt 0" ::: "memory");
    __syncthreads();

    for (int kt = 0; kt < NKT; ++kt) {
        const int buf = kt & 1;
        if (kt + 1 < NKT) asyncLoadTile(kt + 1, buf ^ 1);  // overlap DMA w/ math

#pragma unroll
        for (int ks = 0; ks < BK; ks += 4) {
            const int kk = ks + koff;
            v2f af[4], bf[2];
#pragma unroll
            for (int mt = 0; mt < 4; ++mt)
                af[mt] = *(const v2f*)&ldsA[buf][m0 + mt * 16 + ml][kk];
#pragma unroll
            for (int nt = 0; nt < 2; ++nt)
                bf[nt] = *(const v2f*)&ldsB[buf][n0 + nt * 16 + ml][kk];
#pragma unroll
            for (int mt = 0; mt < 4; ++mt)
#pragma unroll
                for (int nt = 0; nt < 2; ++nt)
                    acc[mt][nt] = __builtin_amdgcn_wmma_f32_16x16x4_f32(
                        false, af[mt], false, bf[nt],
                        (short)0, acc[mt][nt], false, false);
        }

        if (kt + 1 < NKT)
            asm volatile("s_wait_asynccnt 0" ::: "memory");
        __syncthreads();
    }

    // Epilogue: C/D layout — VGPR r: lanes 0-15 hold M=r, lanes 16-31 hold M=r+8.
#pragma unroll
    for (int mt = 0; mt < 4; ++mt) {
#pragma unroll
        for (int nt = 0; nt < 2; ++nt) {
            const int col = col0 + n0 + nt * 16 + ml;
#pragma unroll
            for (int r = 0; r < 8; ++r) {
                const int row = row0 + m0 + mt * 16 + khalf * 8 + r;
                out[(size_t)row * K_OUT + col] = acc[mt][nt][r];
            }
        }
    }
}

extern "C" void kernel_launch(void* const* d_in, const int* in_sizes, int n_in,
                              void* d_out, int out_size, void* d_ws, size_t ws_size,
                              hipStream_t stream) {
    const float* x   = (const float*)d_in[0];   // [N, 1024] fp32
    const int*   esz = (const int*)d_in[1];     // [E] int32
    const float* w   = (const float*)d_in[2];   // [E, 1024, 4096] fp32
    float*       out = (float*)d_out;           // [N, 4096] fp32

    const int E      = in_sizes[1];
    const int n_rows = in_sizes[0] / K_IN;      // total tokens N

    dim3 grid(n_rows / BM, K_OUT / BN);         // 128 x 32 blocks
    dim3 block(256);                            // 8 wave32 waves
    moe_grouped_gemm_f32_wmma<<<grid, block, 0, stream>>>(x, esz, w, out, E);
}